// GNN_15539191677055
// MI455X (gfx1250) — compile-verified
//
#include <hip/hip_runtime.h>

typedef __attribute__((ext_vector_type(2))) float v2f;
typedef __attribute__((ext_vector_type(8))) float v8f;

#define N_NODES 50000
#define N_EDGES 800000
#define HID 128
#define N_GRAPHS 256

// ---------------- zero fill ----------------
__global__ void zero_f32(float* __restrict__ p, long n) {
    long i = (long)blockIdx.x * blockDim.x + threadIdx.x;
    if (i < n) p[i] = 0.0f;
}

// ---------------- weight repack into WMMA-fragment-major order -------------
// Entry (nt, ks, lane) holds the B-fragment pair for both weight matrices:
//   {Wrel[k0][n], Wrel[k0+1][n], Wroot[k0][n], Wroot[k0+1][n]}
// with n = nt*16 + (lane&15), k0 = ks*4 + (lane>>4)*2.
// Inner GEMM loop then needs one coalesced b128 load per (nt, ks).
__global__ __launch_bounds__(256)
void repack_w(const float* __restrict__ Wrel, const float* __restrict__ Wroot,
              float4* __restrict__ Wpk) {
    int t = blockIdx.x * blockDim.x + threadIdx.x;   // 8 nt * 32 ks * 32 lanes
    if (t >= 8 * 32 * 32) return;
    int lane = t & 31;
    int ks   = (t >> 5) & 31;
    int nt   = t >> 10;
    int n  = nt * 16 + (lane & 15);
    int k0 = ks * 4 + (lane >> 4) * 2;
    float4 w;
    w.x = Wrel [(long)k0 * HID + n];
    w.y = Wrel [(long)(k0 + 1) * HID + n];
    w.z = Wroot[(long)k0 * HID + n];
    w.w = Wroot[(long)(k0 + 1) * HID + n];
    Wpk[t] = w;
}

// ---------------- edge scatter-add: agg[dst] += feat[src] ----------------
// 32 threads per edge, 4 floats per thread (float4 gather + 4 f32 HW atomics).
__global__ void scatter_add(const float* __restrict__ feat,
                            const int* __restrict__ src,
                            const int* __restrict__ dst,
                            float* __restrict__ agg, int n_edges) {
    long t = (long)blockIdx.x * blockDim.x + threadIdx.x;
    int edge = (int)(t >> 5);
    int lane = (int)(t & 31);
    if (edge >= n_edges) return;
    int s = src[edge];
    int d = dst[edge];
    const float4 v = *(const float4*)(feat + (long)s * HID + lane * 4);
    float* o = agg + (long)d * HID + lane * 4;
    unsafeAtomicAdd(o + 0, v.x);
    unsafeAtomicAdd(o + 1, v.y);
    unsafeAtomicAdd(o + 2, v.z);
    unsafeAtomicAdd(o + 3, v.w);
}

// ---------------- fused GraphConv GEMM ----------------
// out = relu(Aagg @ Wrel + bias + Aroot @ Wroot), [n_rows,128]x[128,128]
// One wave per 16-row strip, 8 column tiles of 16, K in steps of 4 with
// V_WMMA_F32_16X16X4_F32 (exact fp32). B-fragments come from the repacked
// fragment-major buffer: one global_load_b128 feeds both WMMAs of a tile.
__global__ __launch_bounds__(256)
void gconv_gemm(const float* __restrict__ Aagg, const float* __restrict__ Aroot,
                const float4* __restrict__ Wpk, const float* __restrict__ bias,
                float* __restrict__ out, int n_rows) {
    const int lane = threadIdx.x & 31;
    const int wave = threadIdx.x >> 5;
    const int strip = blockIdx.x * 8 + wave;
    const int m0 = strip * 16;
    if (m0 >= n_rows) return;          // wave-uniform: EXEC stays all-ones for WMMA

    const int mr = lane & 15;          // A: row-in-strip; D: N
    const int kh = lane >> 4;          // K-half selector (K-pair 0/1 vs 2/3)
    const long rowA = (long)(m0 + mr) * HID;
    const float4* __restrict__ wp = Wpk + lane;   // + (nt*32 + ks)*32 per frag

    v8f acc[8] = {};

    for (int ks = 0; ks < 32; ++ks) {
        const int k0 = ks * 4 + kh * 2;
        // A frags: {A[m][k0], A[m][k0+1]}  (8B-aligned contiguous)
        v2f a1 = *(const v2f*)(Aagg  + rowA + k0);
        v2f a2 = *(const v2f*)(Aroot + rowA + k0);
        #pragma unroll
        for (int nt = 0; nt < 8; ++nt) {
            const float4 w = wp[(nt * 32 + ks) * 32];
            v2f b1; b1.x = w.x; b1.y = w.y;
            v2f b2; b2.x = w.z; b2.y = w.w;
            acc[nt] = __builtin_amdgcn_wmma_f32_16x16x4_f32(
                false, a1, false, b1, (short)0, acc[nt], false, false);
            acc[nt] = __builtin_amdgcn_wmma_f32_16x16x4_f32(
                false, a2, false, b2, (short)0, acc[nt], false, false);
        }
    }

    // D layout: VGPR r -> M = r + 8*kh ; N = mr
    #pragma unroll
    for (int nt = 0; nt < 8; ++nt) {
        const int col = nt * 16 + mr;
        const float b = bias[col];
        #pragma unroll
        for (int r = 0; r < 8; ++r) {
            const long row = m0 + r + 8 * kh;
            float v = acc[nt][r] + b;
            out[row * HID + col] = fmaxf(v, 0.0f);
        }
    }
}

// ---------------- segment-sum pool (atomic) ----------------
__global__ void pool_add(const float* __restrict__ h,
                         const int* __restrict__ batch,
                         float* __restrict__ pooled,
                         float* __restrict__ counts, int n_nodes) {
    long t = (long)blockIdx.x * blockDim.x + threadIdx.x;
    int node = (int)(t >> 5);
    int lane = (int)(t & 31);
    if (node >= n_nodes) return;
    int g = batch[node];
    const float4 v = *(const float4*)(h + (long)node * HID + lane * 4);
    float* o = pooled + (long)g * HID + lane * 4;
    unsafeAtomicAdd(o + 0, v.x);
    unsafeAtomicAdd(o + 1, v.y);
    unsafeAtomicAdd(o + 2, v.z);
    unsafeAtomicAdd(o + 3, v.w);
    if (lane == 0) unsafeAtomicAdd(counts + g, 1.0f);
}

// ---------------- MLP head: mean, fc1+relu, fc2 ----------------
__global__ __launch_bounds__(128)
void head(const float* __restrict__ pooled, const float* __restrict__ counts,
          const float* __restrict__ fc1w, const float* __restrict__ fc1b,
          const float* __restrict__ fc2w, const float* __restrict__ fc2b,
          float* __restrict__ out) {
    const int g = blockIdx.x;
    const int j = threadIdx.x;           // 128 threads
    __shared__ float sp[HID];
    __shared__ float sr[HID];
    const float cnt = fmaxf(counts[g], 1.0f);
    sp[j] = pooled[(long)g * HID + j] / cnt;
    __syncthreads();
    float acc = fc1b[j];
    #pragma unroll 8
    for (int c = 0; c < HID; ++c) acc = fmaf(sp[c], fc1w[(long)c * HID + j], acc);
    sr[j] = fmaxf(acc, 0.0f) * fc2w[j];
    __syncthreads();
    for (int s = 64; s > 0; s >>= 1) {
        if (j < s) sr[j] += sr[j + s];
        __syncthreads();
    }
    if (j == 0) out[g] = sr[0] + fc2b[0];
}

extern "C" void kernel_launch(void* const* d_in, const int* in_sizes, int n_in,
                              void* d_out, int out_size, void* d_ws, size_t ws_size,
                              hipStream_t stream) {
    const float* x       = (const float*)d_in[0];
    const int*   ei      = (const int*)d_in[1];     // [2, E]
    const int*   batch   = (const int*)d_in[2];
    const float* w1_rel  = (const float*)d_in[3];
    const float* b1_rel  = (const float*)d_in[4];
    const float* w1_root = (const float*)d_in[5];
    const float* w2_rel  = (const float*)d_in[6];
    const float* b2_rel  = (const float*)d_in[7];
    const float* w2_root = (const float*)d_in[8];
    const float* fc1_w   = (const float*)d_in[9];
    const float* fc1_b   = (const float*)d_in[10];
    const float* fc2_w   = (const float*)d_in[11];
    const float* fc2_b   = (const float*)d_in[12];
    float* out = (float*)d_out;

    const int* src = ei;
    const int* dst = ei + N_EDGES;

    const long feat_elems = (long)N_NODES * HID;     // 6.4M floats
    float*  agg    = (float*)d_ws;
    float*  h1     = agg + feat_elems;
    float*  h2     = h1  + feat_elems;
    float*  pooled = h2  + feat_elems;
    float*  counts = pooled + (long)N_GRAPHS * HID;
    float4* wpk1   = (float4*)(counts + N_GRAPHS);   // 8*32*32 float4 each
    float4* wpk2   = wpk1 + 8 * 32 * 32;

    const int ZB = 256;
    const int zgrid_feat = (int)((feat_elems + ZB - 1) / ZB);
    const int sgrid = (int)(((long)N_EDGES * 32 + ZB - 1) / ZB);
    const int ggrid = (N_NODES / 16 + 7) / 8;        // 3125 strips / 8 waves
    const int pgrid = (int)(((long)N_NODES * 32 + ZB - 1) / ZB);
    const long pool_elems = (long)N_GRAPHS * HID + N_GRAPHS;

    // ---- weight repack (once, reused by every strip) ----
    repack_w<<<(8 * 32 * 32 + ZB - 1) / ZB, ZB, 0, stream>>>(w1_rel, w1_root, wpk1);
    repack_w<<<(8 * 32 * 32 + ZB - 1) / ZB, ZB, 0, stream>>>(w2_rel, w2_root, wpk2);

    // ---- layer 1 ----
    zero_f32<<<zgrid_feat, ZB, 0, stream>>>(agg, feat_elems);
    scatter_add<<<sgrid, ZB, 0, stream>>>(x, src, dst, agg, N_EDGES);
    gconv_gemm<<<ggrid, 256, 0, stream>>>(agg, x, wpk1, b1_rel, h1, N_NODES);
    // ---- layer 2 ----
    zero_f32<<<zgrid_feat, ZB, 0, stream>>>(agg, feat_elems);
    scatter_add<<<sgrid, ZB, 0, stream>>>(h1, src, dst, agg, N_EDGES);
    gconv_gemm<<<ggrid, 256, 0, stream>>>(agg, h1, wpk2, b2_rel, h2, N_NODES);
    // ---- pool + head ----
    zero_f32<<<(int)((pool_elems + ZB - 1) / ZB), ZB, 0, stream>>>(pooled, pool_elems);
    pool_add<<<pgrid, ZB, 0, stream>>>(h2, batch, pooled, counts, N_NODES);
    head<<<N_GRAPHS, HID, 0, stream>>>(pooled, counts, fc1_w, fc1_b, fc2_w, fc2_b, out);
}